// RomHop_batch_16673063043217
// MI455X (gfx1250) — compile-verified
//
#include <hip/hip_runtime.h>

#define B_ 16
#define Q_ 64
#define M_ 512
#define F_ 16
#define D_ 512
#define INF_ 1e20f

typedef float v2f __attribute__((ext_vector_type(2)));
typedef float v8f __attribute__((ext_vector_type(8)));

__device__ __forceinline__ v8f wmma_f32_k4(v2f a, v2f b, v8f c) {
    // V_WMMA_F32_16X16X4_F32 : D = A(16x4) * B(4x16) + C(16x16), all fp32
    return __builtin_amdgcn_wmma_f32_16x16x4_f32(
        false, a, false, b, (short)0, c, false, false);
}

// ---------------------------------------------------------------------------
// K1: out_mem[b,m,d] = sum_f out_me[b,m,f,d]
//     in_mem[b,m,d]  = sum_f softmax_f(mask)[b,m,f] * in_me[b,m,f,d]
// One block per (b,m). Pure bandwidth: streams 512 MB.
// ---------------------------------------------------------------------------
__global__ void k1_reduceF(const float* __restrict__ in_me,
                           const float* __restrict__ out_me,
                           const float* __restrict__ cmask,
                           float* __restrict__ out_mem,
                           float* __restrict__ in_mem) {
    int bm = blockIdx.x;                 // 0 .. B*M-1
    int t  = threadIdx.x;
    __shared__ float raw[F_];
    __shared__ float w[F_];
    if (t < F_) {
        float cm = cmask[bm * F_ + t];
        raw[t] = cm * 1.0f - (1.0f - cm) * INF_;
    }
    __syncthreads();
    if (t < F_) {
        float mx = -INF_;
        #pragma unroll
        for (int f = 0; f < F_; ++f) mx = fmaxf(mx, raw[f]);
        float s = 0.0f;
        #pragma unroll
        for (int f = 0; f < F_; ++f) s += __expf(raw[f] - mx);
        w[t] = __expf(raw[t] - mx) / s;
    }
    __syncthreads();

    const float* ip = in_me  + (size_t)bm * F_ * D_;
    const float* op = out_me + (size_t)bm * F_ * D_;
    for (int d = t; d < D_; d += blockDim.x) {
        float ao = 0.0f, ai = 0.0f;
        #pragma unroll
        for (int f = 0; f < F_; ++f) {
            ao += op[f * D_ + d];
            ai += w[f] * ip[f * D_ + d];
        }
        out_mem[(size_t)bm * D_ + d] = ao;
        in_mem [(size_t)bm * D_ + d] = ai;
    }
}

// ---------------------------------------------------------------------------
// K2: att_qm[b,q,m] = max_f( masked( query[b,q,:] . in_me[b,m,f,:] ) )
// One wave per (b,m). N-dim of each WMMA tile == F == 16, so a tile is
// (16 q) x (16 f) for a single m; in_memory_embed is read exactly once.
// ---------------------------------------------------------------------------
__global__ void __launch_bounds__(256)
k2_attention(const float* __restrict__ query,
             const float* __restrict__ in_me,
             const float* __restrict__ cmask,
             const float* __restrict__ qmask,
             float* __restrict__ att) {
    int wave = threadIdx.x >> 5;
    int lane = threadIdx.x & 31;
    int blk  = blockIdx.x;               // 1024 blocks
    int b    = blk >> 6;                 // 64 blocks per batch
    int m    = ((blk & 63) << 3) + wave; // 8 waves -> 8 m's per block
    int half = lane >> 4;
    int lm   = lane & 15;

    // B tile: lane holds B[k = 2*half+v][n = lm] = in_me[b,m,f=lm,d0+2*half+v]
    const float* bp  = in_me + (((size_t)(b * M_ + m)) * F_ + lm) * D_ + 2 * half;
    // A tile(s): lane holds A[row = lm][k = 2*half+v] = query[b,q0+lm,d0+2*half+v]
    const float* ap0 = query + ((size_t)(b * Q_ + lm)) * D_ + 2 * half;

    v8f acc[4] = {v8f{}, v8f{}, v8f{}, v8f{}};
    #pragma unroll 4
    for (int d0 = 0; d0 < D_; d0 += 4) {
        v2f bv = *(const v2f*)(bp + d0);
        #pragma unroll
        for (int t = 0; t < 4; ++t) {
            v2f av = *(const v2f*)(ap0 + (size_t)t * 16 * D_ + d0);
            acc[t] = wmma_f32_k4(av, bv, acc[t]);
        }
    }

    float cm = cmask[(b * M_ + m) * F_ + lm];       // per-lane f = lm
    const float* qm = qmask + b * Q_;
    float* ao = att + ((size_t)b * Q_) * M_ + m;

    #pragma unroll
    for (int t = 0; t < 4; ++t) {
        #pragma unroll
        for (int r = 0; r < 8; ++r) {
            int q  = 16 * t + r + 8 * half;          // C row for this lane-half
            float v = acc[t][r];
            v = cm * v - (1.0f - cm) * INF_;
            float qv = qm[q];
            v = qv * v - (1.0f - qv) * INF_;
            // max over the 16 columns (f) held in this 16-lane half
            #pragma unroll
            for (int off = 8; off >= 1; off >>= 1)
                v = fmaxf(v, __shfl_xor(v, off, 32));
            if (lm == 0) ao[(size_t)q * M_] = v;     // lane 0 and lane 16 write
        }
    }
}

// ---------------------------------------------------------------------------
// K3: probs[b,q,:] = softmax over M. One wave per (b,q) row (512 elems).
// ---------------------------------------------------------------------------
__global__ void k3_softmax_m(const float* __restrict__ att,
                             float* __restrict__ probs) {
    int row  = blockIdx.x * (blockDim.x >> 5) + (threadIdx.x >> 5); // 0..1023
    int lane = threadIdx.x & 31;
    const float* p = att + (size_t)row * M_;
    float vals[16];
    float mx = -INF_;
    #pragma unroll
    for (int i = 0; i < 16; ++i) { vals[i] = p[lane + 32 * i]; mx = fmaxf(mx, vals[i]); }
    #pragma unroll
    for (int off = 16; off >= 1; off >>= 1) mx = fmaxf(mx, __shfl_xor(mx, off, 32));
    float s = 0.0f;
    #pragma unroll
    for (int i = 0; i < 16; ++i) { vals[i] = __expf(vals[i] - mx); s += vals[i]; }
    #pragma unroll
    for (int off = 16; off >= 1; off >>= 1) s += __shfl_xor(s, off, 32);
    float inv = 1.0f / s;
    #pragma unroll
    for (int i = 0; i < 16; ++i) probs[(size_t)row * M_ + lane + 32 * i] = vals[i] * inv;
}

// ---------------------------------------------------------------------------
// K4: new_query[b,q,d] = query[b,q,d] + sum_m probs[b,q,m] * out_mem[b,m,d]
// One wave per 16x16 output tile, K = M = 512.
// ---------------------------------------------------------------------------
__global__ void __launch_bounds__(256)
k4_newquery(const float* __restrict__ probs,
            const float* __restrict__ out_mem,
            const float* __restrict__ query,
            float* __restrict__ nq) {
    int wg   = blockIdx.x * (blockDim.x >> 5) + (threadIdx.x >> 5); // 0..2047
    int lane = threadIdx.x & 31;
    int half = lane >> 4, lm = lane & 15;
    int b    = wg >> 7;               // 128 tiles per batch
    int rest = wg & 127;
    int q0   = (rest >> 5) << 4;      // 4 q-tiles
    int d0   = (rest & 31) << 4;      // 32 d-tiles

    const float* ap = probs   + ((size_t)(b * Q_ + q0 + lm)) * M_ + 2 * half;
    const float* bp = out_mem + ((size_t)(b * M_ + 2 * half)) * D_ + d0 + lm;

    v8f acc = {};
    #pragma unroll 4
    for (int k0 = 0; k0 < M_; k0 += 4) {
        v2f a = *(const v2f*)(ap + k0);
        v2f bv;
        bv.x = bp[(size_t)(k0    ) * D_];
        bv.y = bp[(size_t)(k0 + 1) * D_];
        acc = wmma_f32_k4(a, bv, acc);
    }
    #pragma unroll
    for (int r = 0; r < 8; ++r) {
        int q = q0 + r + 8 * half;
        size_t o = ((size_t)(b * Q_ + q)) * D_ + d0 + lm;
        nq[o] = query[o] + acc[r];
    }
}

// ---------------------------------------------------------------------------
// K5: probs2[b,q,m] = softmax over Q. One thread per (b,m) column.
// ---------------------------------------------------------------------------
__global__ void k5_softmax_q(const float* __restrict__ att,
                             float* __restrict__ probs2) {
    int idx = blockIdx.x * blockDim.x + threadIdx.x;   // 0 .. B*M-1
    if (idx >= B_ * M_) return;
    int b = idx / M_, m = idx % M_;
    const float* p = att + ((size_t)b * Q_) * M_ + m;
    float mx = -INF_;
    for (int q = 0; q < Q_; ++q) mx = fmaxf(mx, p[(size_t)q * M_]);
    float s = 0.0f;
    for (int q = 0; q < Q_; ++q) s += __expf(p[(size_t)q * M_] - mx);
    float inv = 1.0f / s;
    float* o = probs2 + ((size_t)b * Q_) * M_ + m;
    for (int q = 0; q < Q_; ++q) o[(size_t)q * M_] = __expf(p[(size_t)q * M_] - mx) * inv;
}

// ---------------------------------------------------------------------------
// K6: in_mem[b,m,d] += sum_q probs2[b,q,m] * new_query[b,q,d]
// One wave per 16x16 output tile, K = Q = 64.
// ---------------------------------------------------------------------------
__global__ void __launch_bounds__(256)
k6_inmem(const float* __restrict__ probs2,
         const float* __restrict__ nq,
         float* __restrict__ in_mem) {
    int wg   = blockIdx.x * (blockDim.x >> 5) + (threadIdx.x >> 5); // 0..16383
    int lane = threadIdx.x & 31;
    int half = lane >> 4, lm = lane & 15;
    int b    = wg >> 10;              // 1024 tiles per batch
    int rest = wg & 1023;
    int m0   = (rest >> 5) << 4;
    int d0   = (rest & 31) << 4;

    // A[row=lm][k] = probs2[b, q=k0+2*half+v, m0+lm]
    const float* ap = probs2 + ((size_t)(b * Q_ + 2 * half)) * M_ + m0 + lm;
    // B[k][n=lm]   = nq[b, q=k0+2*half+v, d0+lm]
    const float* bp = nq     + ((size_t)(b * Q_ + 2 * half)) * D_ + d0 + lm;

    v8f acc = {};
    #pragma unroll
    for (int k0 = 0; k0 < Q_; k0 += 4) {
        v2f a, bv;
        a.x  = ap[(size_t)(k0    ) * M_];
        a.y  = ap[(size_t)(k0 + 1) * M_];
        bv.x = bp[(size_t)(k0    ) * D_];
        bv.y = bp[(size_t)(k0 + 1) * D_];
        acc = wmma_f32_k4(a, bv, acc);
    }
    #pragma unroll
    for (int r = 0; r < 8; ++r) {
        int m = m0 + r + 8 * half;
        size_t o = ((size_t)(b * M_ + m)) * D_ + d0 + lm;
        in_mem[o] += acc[r];
    }
}

// ---------------------------------------------------------------------------
extern "C" void kernel_launch(void* const* d_in, const int* in_sizes, int n_in,
                              void* d_out, int out_size, void* d_ws, size_t ws_size,
                              hipStream_t stream) {
    const float* query  = (const float*)d_in[0];   // (B,Q,D)
    const float* in_me  = (const float*)d_in[1];   // (B,M,F,D)
    const float* out_me = (const float*)d_in[2];   // (B,M,F,D)
    const float* cmask  = (const float*)d_in[3];   // (B,M,F)
    const float* qmask  = (const float*)d_in[4];   // (B,Q)

    // d_out: [ new_query (B*Q*D) | in_mem (B*M*D) | out_mem (B*M*D) ]
    float* nq      = (float*)d_out;
    float* in_mem  = nq + (size_t)B_ * Q_ * D_;
    float* out_mem = in_mem + (size_t)B_ * M_ * D_;

    // workspace: att_qm | probs | probs2, each B*Q*M floats (2 MB each)
    float* att    = (float*)d_ws;
    float* probs  = att + (size_t)B_ * Q_ * M_;
    float* probs2 = probs + (size_t)B_ * Q_ * M_;

    // K1: F reductions (also initializes in_mem/out_mem outputs)
    k1_reduceF<<<B_ * M_, 256, 0, stream>>>(in_me, out_me, cmask, out_mem, in_mem);
    // K2: WMMA attention + masks + max over F
    k2_attention<<<(B_ * M_) / 8, 256, 0, stream>>>(query, in_me, cmask, qmask, att);
    // K3: softmax over M
    k3_softmax_m<<<(B_ * Q_) / 8, 256, 0, stream>>>(att, probs);
    // K4: WMMA new_query = query + probs @ out_mem
    k4_newquery<<<(B_ * (Q_ / 16) * (D_ / 16)) / 8, 256, 0, stream>>>(probs, out_mem, query, nq);
    // K5: softmax over Q
    k5_softmax_q<<<(B_ * M_) / 256, 256, 0, stream>>>(att, probs2);
    // K6: WMMA in_mem += probs2^T @ new_query
    k6_inmem<<<(B_ * (M_ / 16) * (D_ / 16)) / 8, 256, 0, stream>>>(probs2, nq, in_mem);
}